// R_TuckER_6201932775560
// MI455X (gfx1250) — compile-verified
//
#include <hip/hip_runtime.h>
#include <hip/hip_bf16.h>
#include <cmath>

// TuckER: out = sigmoid( ((r (x) e1) @ reshape(W, [D*D, D])) @ E2^T )
// D=200, B=512, N_ENT=100000. All fp32, WMMA f32 16x16x4.

#define DD      200
#define DP      204           // padded LDS row stride (dwords): 16-lane strided reads hit 16 distinct banks
#define NENT    100000
#define BATCH   512
#define KSLICES 8             // K-split of the core GEMM across blockIdx.z
#define ISLICE  (DD / KSLICES) // 25 i-planes per slice

typedef float v2f __attribute__((ext_vector_type(2)));
typedef float v8f __attribute__((ext_vector_type(8)));

static __device__ __forceinline__ v8f wmma_f32(v2f a, v2f b, v8f c) {
    return __builtin_amdgcn_wmma_f32_16x16x4_f32(
        /*neg_a=*/false, a, /*neg_b=*/false, b,
        /*c_mod=*/(short)0, c, /*reuse_a=*/false, /*reuse_b=*/false);
}

// ---------------------------------------------------------------------------
// Kernel 1: part[ks][b, n] = sum_{i in slice ks, j} r[b,i]*e1[b,j]*W[i,j,n]
// GEMM (512 x 40000) @ (40000 x 200); A generated on the fly from LDS tiles.
// One wave computes TWO 16x16 tiles of x (shared B fragments, two independent
// WMMA accumulator chains). Grid = (16 M-pairs, 13 N-tiles, 8 K-slices).
// ---------------------------------------------------------------------------
__global__ __launch_bounds__(32) void tucker_core_gemm(
    const int* __restrict__ X, const float* __restrict__ E1,
    const float* __restrict__ R, const float* __restrict__ W,
    float* __restrict__ part)          // [KSLICES][BATCH][DD]
{
    __shared__ float r_lds [32 * DP];  // 26112 B
    __shared__ float e1_lds[32 * DP];  // 26112 B

    const int mt2  = blockIdx.x;       // 0..15  pair of batch tiles (rows mt2*32..+31)
    const int nt   = blockIdx.y;       // 0..12  output-col tile (last partial)
    const int ks   = blockIdx.z;       // 0..7   K slice (i-plane range)
    const int lane = threadIdx.x;      // 0..31
    const int b0   = mt2 * 32;
    const int n0   = nt * 16;
    const int l16  = lane & 15;
    const int hk   = lane >> 4;

    // Gather the 32 e1 / r rows for this batch pair into LDS.
    for (int idx = lane; idx < 32 * DD; idx += 32) {
        const int row = idx / DD, col = idx % DD;
        const int b    = b0 + row;
        const int subj = X[2 * b + 0];
        const int rel  = X[2 * b + 1];
        r_lds [row * DP + col] = R [(size_t)rel  * DD + col];
        e1_lds[row * DP + col] = E1[(size_t)subj * DD + col];
    }
    __syncthreads();

    const int nIdx   = n0 + l16;
    const int nClamp = nIdx < DD ? nIdx : (DD - 1);      // keep OOB lanes in-bounds
    const float* __restrict__ Wcol = W + nClamp;         // column base: W[i,j,nClamp]

    v8f c0 = {0.f, 0.f, 0.f, 0.f, 0.f, 0.f, 0.f, 0.f};  // tile 0 (rows b0..b0+15)
    v8f c1 = {0.f, 0.f, 0.f, 0.f, 0.f, 0.f, 0.f, 0.f};  // tile 1 (rows b0+16..b0+31)

    const float* __restrict__ e1r0 = &e1_lds[l16 * DP];
    const float* __restrict__ e1r1 = &e1_lds[(16 + l16) * DP];

    const int i_begin = ks * ISLICE, i_end = i_begin + ISLICE;
    for (int i = i_begin; i < i_end; ++i) {
        const float rv0 = r_lds[l16 * DP + i];
        const float rv1 = r_lds[(16 + l16) * DP + i];
        const float* __restrict__ Wrow = Wcol + (size_t)i * (DD * DD); // W[i, j, n]
        if (i + 1 < i_end)
            __builtin_prefetch(Wcol + (size_t)(i + 1) * (DD * DD), 0, 0);

        for (int j0 = 0; j0 < DD; j0 += 4) {
            const int j = j0 + 2 * hk;                   // this lane-half's K pair
            v2f b;                                       // shared by both M tiles
            b.x = Wrow[(size_t)j       * DD];
            b.y = Wrow[(size_t)(j + 1) * DD];
            v2f a0; a0.x = rv0 * e1r0[j]; a0.y = rv0 * e1r0[j + 1];
            v2f a1; a1.x = rv1 * e1r1[j]; a1.y = rv1 * e1r1[j + 1];
            c0 = wmma_f32(a0, b, c0);                    // two independent chains
            c1 = wmma_f32(a1, b, c1);
        }
    }

    // C/D layout: VGPR v -> row M = v + 8*hk, col N = l16
    if (nIdx < DD) {
        float* __restrict__ dst = part + (size_t)ks * (BATCH * DD);
        for (int v = 0; v < 8; ++v) {
            const int m = b0 + v + 8 * hk;
            dst[(size_t)m        * DD + nIdx] = c0[v];
            dst[(size_t)(m + 16) * DD + nIdx] = c1[v];
        }
    }
}

// ---------------------------------------------------------------------------
// Kernel 2: out[b, n] = sigmoid( sum_k x[b,k] * E2[n,k] ), x = sum of partials.
// Block = 256 threads (8 waves). Each block owns 128 batch rows (x reduced
// into LDS once) and grid-strides over 16-wide N tiles; each E2 tile staged in
// LDS, shared by all 8 waves; next tile prefetched during compute.
// ---------------------------------------------------------------------------
__global__ __launch_bounds__(256) void tucker_score_gemm(
    const float* __restrict__ part, const float* __restrict__ E2,
    float* __restrict__ out)
{
    extern __shared__ float smem[];
    float* x_lds  = smem;                 // 128 * DP
    float* e2_lds = smem + 128 * DP;      // 16  * DP

    const int mg     = blockIdx.x;        // 0..3  (128 rows each)
    const int m_base = mg * 128;
    const int tid    = threadIdx.x;
    const int wave   = tid >> 5;          // 0..7
    const int lane   = tid & 31;
    const int l16    = lane & 15;
    const int hk     = lane >> 4;

    // Stage this block's 128 rows of x into LDS, reducing the 8 K-slices.
    for (int idx = tid; idx < 128 * DD; idx += 256) {
        const int row = idx / DD, col = idx % DD;
        const size_t off = (size_t)(m_base + row) * DD + col;
        float s = 0.f;
        #pragma unroll
        for (int ksl = 0; ksl < KSLICES; ++ksl)
            s += part[(size_t)ksl * (BATCH * DD) + off];
        x_lds[row * DP + col] = s;
    }

    const float* __restrict__ xrow = &x_lds[(wave * 16 + l16) * DP]; // A row (M=l16 of wave's tile)
    const float* __restrict__ e2c  = &e2_lds[l16 * DP];              // B col n=l16 -> E2 row

    const int NTILES = NENT / 16;         // 6250
    for (int nt = blockIdx.y; nt < NTILES; nt += gridDim.y) {
        const int n0 = nt * 16;

        __syncthreads();                  // prior tile fully consumed (also covers x_lds)
        for (int idx = tid; idx < 16 * DD; idx += 256) {
            const int row = idx / DD, col = idx % DD;
            e2_lds[row * DP + col] = E2[(size_t)(n0 + row) * DD + col];
        }
        __syncthreads();

        // Prefetch next E2 tile (16 rows x 200 f32 = contiguous 12.8 KB,
        // one prefetch per 64B line) to overlap with the WMMA phase.
        const int n_next = n0 + 16 * (int)gridDim.y;
        if (n_next < NENT && tid < (16 * DD) / 16)       // 200 lines
            __builtin_prefetch(E2 + (size_t)n_next * DD + (size_t)tid * 16, 0, 0);

        v8f c0 = {0.f, 0.f, 0.f, 0.f, 0.f, 0.f, 0.f, 0.f};
        v8f c1 = {0.f, 0.f, 0.f, 0.f, 0.f, 0.f, 0.f, 0.f};
        for (int k0 = 0; k0 < DD; k0 += 8) {             // 25 iters, 2 indep chains
            const int ka = k0 + 2 * hk;
            v2f a, b;
            a.x = xrow[ka]; a.y = xrow[ka + 1];
            b.x = e2c[ka];  b.y = e2c[ka + 1];
            c0 = wmma_f32(a, b, c0);
            const int kb = k0 + 4 + 2 * hk;
            a.x = xrow[kb]; a.y = xrow[kb + 1];
            b.x = e2c[kb];  b.y = e2c[kb + 1];
            c1 = wmma_f32(a, b, c1);
        }

        // Fused sigmoid + store. Row m = wave tile row v + 8*hk, col n0+l16.
        const int n = n0 + l16;
        for (int v = 0; v < 8; ++v) {
            const int m = m_base + wave * 16 + v + 8 * hk;
            const float val = c0[v] + c1[v];
            out[(size_t)m * NENT + n] = 1.0f / (1.0f + expf(-val));
        }
    }
}

// ---------------------------------------------------------------------------
extern "C" void kernel_launch(void* const* d_in, const int* in_sizes, int n_in,
                              void* d_out, int out_size, void* d_ws, size_t ws_size,
                              hipStream_t stream) {
    const int*   X  = (const int*)  d_in[0];   // (512, 2)
    const float* E1 = (const float*)d_in[1];   // (100000, 200)
    const float* R  = (const float*)d_in[2];   // (500, 200)
    const float* E2 = (const float*)d_in[3];   // (100000, 200)
    const float* W  = (const float*)d_in[4];   // (200, 200, 200)
    float* out  = (float*)d_out;               // (512, 100000)
    float* part = (float*)d_ws;                // [8][512][200] partials (3.2 MB)

    // Phase 1: fused gather + (r (x) e1) @ W2 -> 8 K-slice partials of x
    dim3 g1(BATCH / 32, (DD + 15) / 16, KSLICES);   // 16 x 13 x 8
    tucker_core_gemm<<<g1, 32, 0, stream>>>(X, E1, R, W, part);

    // Phase 2: reduce partials, x @ E2^T + sigmoid -> out (512 x 100000)
    dim3 g2(BATCH / 128, 625);                      // 4 x 625, each block does 10 N tiles
    size_t shmem = (size_t)(128 * DP + 16 * DP) * sizeof(float); // ~117.5 KB dynamic LDS
    tucker_score_gemm<<<g2, 256, shmem, stream>>>(part, E2, out);
}